// GraphGatedAttentionLayer_90220083020122
// MI455X (gfx1250) — compile-verified
//
#include <hip/hip_runtime.h>
#include <hip/hip_bf16.h>
#include <math.h>
#include <stdint.h>

// Problem sizes (fixed by the reference)
#define NN    2048
#define IN_F  1024
#define OUT_F 64

// Kernel-2 tiling: 4 node-rows per block, adj streamed in 512-col chunks
#define ROWS_PB   4
#define JCHUNK    512
#define NCHUNKS   (NN / JCHUNK)

typedef float v2f __attribute__((ext_vector_type(2)));
typedef float v8f __attribute__((ext_vector_type(8)));

typedef unsigned int u32x4 __attribute__((ext_vector_type(4)));
typedef int          i32x4 __attribute__((ext_vector_type(4)));
typedef int          i32x8 __attribute__((ext_vector_type(8)));

// -----------------------------------------------------------------------------
// Kernel 1: h = x @ W  (fp32 WMMA 16x16x4), then fused h1 = h@a1, h2 = h@a2.
// One block = 16 rows of x. 4 waves, each owns one 16-col tile of the output.
// (emits 8x-unrolled v_wmma_f32_16x16x4_f32, verified in round 1)
// -----------------------------------------------------------------------------
__global__ __launch_bounds__(128)
void gat_gemm_kernel(const float* __restrict__ x,
                     const float* __restrict__ W,
                     const float* __restrict__ a1,
                     const float* __restrict__ a2,
                     float* __restrict__ h,
                     float* __restrict__ h1,
                     float* __restrict__ h2) {
    __shared__ float hs[16 * OUT_F];          // 16x64 slab of h, 4 KB

    const int wave  = threadIdx.x >> 5;       // 0..3  -> output col tile
    const int lane  = threadIdx.x & 31;
    const int m     = lane & 15;              // M row (A) / N col (B,C,D)
    const int khalf = (lane >> 4) << 1;       // 0 or 2: K offset of this half-wave
    const int row0  = blockIdx.x * 16;
    const int n0    = wave * 16;
    const int n     = n0 + m;                 // global output column for B/C/D

    // ---- Phase 1: 16x16 tile of h = x@W, K = 1024 in steps of 4 ----
    v8f acc = {};
    const float* xrow = x + (size_t)(row0 + m) * IN_F;
    for (int kb = 0; kb < IN_F; kb += 4) {
        v2f a, b;
        a.x = xrow[kb + khalf];
        a.y = xrow[kb + khalf + 1];
        b.x = W[(size_t)(kb + khalf)     * OUT_F + n];
        b.y = W[(size_t)(kb + khalf + 1) * OUT_F + n];
        acc = __builtin_amdgcn_wmma_f32_16x16x4_f32(
                  false, a, false, b, (short)0, acc, false, false);
    }

    // C/D layout: VGPR r, lane half hh -> row (r + hh*8), col n0 + (lane&15)
    const int mbase = (lane >> 4) * 8;
#pragma unroll
    for (int r = 0; r < 8; ++r) {
        const int mm = mbase + r;
        hs[mm * OUT_F + n] = acc[r];
        h[(size_t)(row0 + mm) * OUT_F + n] = acc[r];
    }
    __syncthreads();

    // ---- Phase 2: h1/h2 tiles from the LDS slab, K = 64 in steps of 4 ----
    v8f acc1 = {};
    v8f acc2 = {};
    for (int kb = 0; kb < OUT_F; kb += 4) {
        v2f a, b1, b2;
        a.x = hs[m * OUT_F + kb + khalf];
        a.y = hs[m * OUT_F + kb + khalf + 1];
        b1.x = a1[(kb + khalf)     * OUT_F + n];
        b1.y = a1[(kb + khalf + 1) * OUT_F + n];
        b2.x = a2[(kb + khalf)     * OUT_F + n];
        b2.y = a2[(kb + khalf + 1) * OUT_F + n];
        acc1 = __builtin_amdgcn_wmma_f32_16x16x4_f32(false, a, false, b1,
                                                     (short)0, acc1, false, false);
        acc2 = __builtin_amdgcn_wmma_f32_16x16x4_f32(false, a, false, b2,
                                                     (short)0, acc2, false, false);
    }
#pragma unroll
    for (int r = 0; r < 8; ++r) {
        const int mm = mbase + r;
        h1[(size_t)(row0 + mm) * OUT_F + n] = acc1[r];
        h2[(size_t)(row0 + mm) * OUT_F + n] = acc2[r];
    }
}

// -----------------------------------------------------------------------------
// TDM helper: DMA a ROWS_PB x JCHUNK fp32 tile of adj into LDS.
// D# per CDNA5 ISA §8.3/8.4:
//   group0: [1:0]=count=1, [63:32]=lds_addr, [120:64]=global byte addr,
//           [127:126]=type=2
//   group1: [17:16]=data_size=2 (4B), [79:48]=tensor_dim0=2048,
//           [111:80]=tensor_dim1=2048, [127:112]=tile_dim0=JCHUNK,
//           [143:128]=tile_dim1=ROWS_PB, [207:160]=tensor_dim0_stride=2048
//   groups 2/3: zero (2-D tensor, no iterate/gather)
// LDS result is compacted: lds[r*JCHUNK + c] = adj[(i0+r)*NN + j0 + c]
// This toolchain's builtin is the 6-arg form:
//   (u32x4, i32x8, i32x4, i32x4, i32x8, i32 cpol)
// -----------------------------------------------------------------------------
__device__ __forceinline__
void tdm_load_adj_chunk(const float* gptr, unsigned lds_byte_off) {
    const unsigned long long ga = (unsigned long long)(uintptr_t)gptr;
    u32x4 g0;
    g0[0] = 1u;                                           // count=1, user mode
    g0[1] = lds_byte_off;                                 // lds_addr
    g0[2] = (unsigned)(ga & 0xFFFFFFFFu);                 // global_addr[31:0]
    g0[3] = (unsigned)((ga >> 32) & 0x01FFFFFFu)          // global_addr[56:32]
          | (2u << 30);                                   // type=2 ("image")
    i32x8 g1;
    g1[0] = (int)(2u << 16);                              // data_size = 4 bytes
    g1[1] = (int)((unsigned)(NN & 0xFFFF) << 16);         // tensor_dim0[15:0]
    g1[2] = (int)((unsigned)(NN & 0xFFFF) << 16);         // tensor_dim1[15:0]
    g1[3] = (int)((unsigned)JCHUNK << 16);                // tile_dim0
    g1[4] = ROWS_PB;                                      // tile_dim1
    g1[5] = NN;                                           // tensor_dim0_stride
    g1[6] = 0;
    g1[7] = 0;
    i32x4 g2 = {0, 0, 0, 0};
    i32x4 g3 = {0, 0, 0, 0};
    i32x8 g4 = {0, 0, 0, 0, 0, 0, 0, 0};
    __builtin_amdgcn_tensor_load_to_lds(g0, g1, g2, g3, g4, /*cpol=*/0);
}

// -----------------------------------------------------------------------------
// Kernel 2: h_prime[i,f] = sum_j adj[i,j] * sigmoid(h1[i,f]+h2[j,f]) * h[j,f]
// then out = elu(h_prime).
// blockDim = (64, ROWS_PB): tx = f channel, ty = row-within-block.
// adj is the only HBM-scale stream (16 MB); it is staged into LDS in 8 KB
// chunks by the Tensor Data Mover, double-buffered: wave 0 issues the next
// chunk, s_wait_tensorcnt(1) guarantees the current chunk landed (per-wave
// TDM ops complete in order), then the block scans it from LDS.
// adj[i,j] is wave-uniform (a wave spans 32 f-channels of one row), so the
// a==0 case (~98%) skips the transcendental with a scalar branch.
// -----------------------------------------------------------------------------
__global__ __launch_bounds__(256)
void gat_gate_agg_kernel(const float* __restrict__ adj,
                         const float* __restrict__ h,
                         const float* __restrict__ h1,
                         const float* __restrict__ h2,
                         float* __restrict__ out) {
    __shared__ float adjs[2][ROWS_PB * JCHUNK];           // 2 x 8 KB

    const int f   = threadIdx.x;                          // 0..63
    const int ty  = threadIdx.y;                          // 0..ROWS_PB-1
    const int i0  = blockIdx.x * ROWS_PB;
    const int i   = i0 + ty;
    const int tid = ty * 64 + f;

    const float h1if = h1[(size_t)i * OUT_F + f];

    // Prefetch chunk 0
    if (tid == 0) {
        tdm_load_adj_chunk(adj + (size_t)i0 * NN,
                           (unsigned)(uintptr_t)&adjs[0][0]);
    }

    float acc = 0.0f;
    for (int c = 0; c < NCHUNKS; ++c) {
        // Issue next chunk, then wait for the current one (in-order TDM).
        if (tid == 0) {
            if (c + 1 < NCHUNKS) {
                tdm_load_adj_chunk(adj + (size_t)i0 * NN + (c + 1) * JCHUNK,
                                   (unsigned)(uintptr_t)&adjs[(c + 1) & 1][0]);
                __builtin_amdgcn_s_wait_tensorcnt(1);
            } else {
                __builtin_amdgcn_s_wait_tensorcnt(0);
            }
        }
        __syncthreads();

        const float* arow = &adjs[c & 1][ty * JCHUNK];
        const int    jb   = c * JCHUNK;
        for (int jc = 0; jc < JCHUNK; ++jc) {
            const float a = arow[jc];                     // LDS broadcast
            if (a != 0.0f) {
                const int   j = jb + jc;
                const float z = h1if + h2[(size_t)j * OUT_F + f];
                const float g = 1.0f / (1.0f + __expf(-z));   // sigmoid
                acc = fmaf(a * g, h[(size_t)j * OUT_F + f], acc);
            }
        }
        __syncthreads();                                  // buffer reuse fence
    }

    // ELU (alpha = 1)
    out[(size_t)i * OUT_F + f] = (acc > 0.0f) ? acc : (__expf(acc) - 1.0f);
}

// -----------------------------------------------------------------------------
// Launch: inputs in setup_inputs() order: x, adj, W, a1, a2 (all fp32).
// Workspace layout: h | h1 | h2, each 2048*64 floats (1.5 MB total).
// -----------------------------------------------------------------------------
extern "C" void kernel_launch(void* const* d_in, const int* in_sizes, int n_in,
                              void* d_out, int out_size, void* d_ws, size_t ws_size,
                              hipStream_t stream) {
    const float* x   = (const float*)d_in[0];
    const float* adj = (const float*)d_in[1];
    const float* W   = (const float*)d_in[2];
    const float* a1  = (const float*)d_in[3];
    const float* a2  = (const float*)d_in[4];
    float* out = (float*)d_out;

    float* h  = (float*)d_ws;
    float* h1 = h  + (size_t)NN * OUT_F;
    float* h2 = h1 + (size_t)NN * OUT_F;

    gat_gemm_kernel<<<NN / 16, 128, 0, stream>>>(x, W, a1, a2, h, h1, h2);
    gat_gate_agg_kernel<<<NN / ROWS_PB, dim3(64, ROWS_PB), 0, stream>>>(
        adj, h, h1, h2, out);
}